// BasicGCN_78005196030504
// MI455X (gfx1250) — compile-verified
//
#include <hip/hip_runtime.h>
#include <hip/hip_bf16.h>
#include <stdint.h>

typedef __attribute__((ext_vector_type(2))) float v2f;
typedef __attribute__((ext_vector_type(8))) float v8f;

// ---------------------------------------------------------------------------
// Native f32 atomic add (gfx1250 has GLOBAL_ATOMIC_ADD_F32)
// ---------------------------------------------------------------------------
__device__ __forceinline__ void atomic_add_f32(float* p, float v) {
#if __has_builtin(__builtin_amdgcn_global_atomic_fadd_f32)
    typedef __attribute__((address_space(1))) float gfloat;
    (void)__builtin_amdgcn_global_atomic_fadd_f32((gfloat*)(uintptr_t)p, v);
#else
    (void)__hip_atomic_fetch_add(p, v, __ATOMIC_RELAXED, __HIP_MEMORY_SCOPE_AGENT);
#endif
}

// ---------------------------------------------------------------------------
// Degree / normalization
// ---------------------------------------------------------------------------
__global__ __launch_bounds__(256) void deg_init_kernel(float* deg, int n) {
    int i = blockIdx.x * 256 + threadIdx.x;
    if (i < n) deg[i] = 1.0f;                       // self-loop contribution
}

__global__ __launch_bounds__(256) void deg_accum_kernel(const int* __restrict__ dst,
                                                        float* __restrict__ deg, int E) {
    int e = blockIdx.x * 256 + threadIdx.x;
    if (e < E) atomic_add_f32(deg + dst[e], 1.0f);
}

__global__ __launch_bounds__(256) void dinv_kernel(const float* __restrict__ deg,
                                                   float* __restrict__ dinv, int n) {
    int i = blockIdx.x * 256 + threadIdx.x;
    if (i < n) dinv[i] = rsqrtf(deg[i]);
}

// ---------------------------------------------------------------------------
// WMMA GEMM: Out[n,128] = (RELU ? relu(A) : A)[n,128] @ W[128,128]
// Block = 256 threads (8 waves), each wave owns a 16x128 strip.
// W staged in LDS pre-swizzled into V_WMMA_F32_16X16X4_F32 B-fragment order:
// pair (kt,nt,lane) -> { W[4kt+2*(lane>>4)][16nt+(lane&15)], W[..+1][..] }
// Tail rows: address-clamped only (no zeroing needed -- garbage rows of D are
// simply not stored), so the hot loop has no EXEC manipulation at all.
// ---------------------------------------------------------------------------
template <bool RELU>
__global__ __launch_bounds__(256)
void gcn_gemm128_kernel(const float* __restrict__ A, const float* __restrict__ W,
                        float* __restrict__ Out, int n) {
    __shared__ v2f Wsw[32 * 8 * 32];                // 64 KB

    const int tid = threadIdx.x;
    for (int p = tid; p < 32 * 8 * 32; p += 256) {
        int kt   = p >> 8;
        int nt   = (p >> 5) & 7;
        int lane = p & 31;
        int k = kt * 4 + ((lane >> 4) << 1);
        int c = nt * 16 + (lane & 15);
        v2f w;
        w.x = W[k * 128 + c];
        w.y = W[(k + 1) * 128 + c];
        Wsw[p] = w;
    }
    __syncthreads();

    const int wave    = tid >> 5;
    const int lane    = tid & 31;
    const int halfsel = lane >> 4;                  // 0: lanes 0-15, 1: 16-31
    const int mlane   = lane & 15;
    const long blockBase = (long)blockIdx.x * 128 + wave * 16;
    long row = blockBase + mlane;                   // A row for this lane
    if (row > (long)n - 1) row = (long)n - 1;       // clamp: memory-safe, rows >= n unused
    const float* arow = A + row * 128 + halfsel * 2;

    v8f acc[8] = {};

#pragma unroll 4
    for (int kt = 0; kt < 32; ++kt) {
        v2f a = *(const v2f*)(arow + kt * 4);
        if (RELU) {
            a.x = fmaxf(a.x, 0.0f);
            a.y = fmaxf(a.y, 0.0f);
        }
#pragma unroll
        for (int nt = 0; nt < 8; ++nt) {
            v2f b = Wsw[(kt << 8) + (nt << 5) + lane];
            acc[nt] = __builtin_amdgcn_wmma_f32_16x16x4_f32(
                false, a, false, b, (short)0, acc[nt], false, false);
        }
    }

    // D layout: VGPR r -> M = r + 8*halfsel, N = nt*16 + mlane
#pragma unroll
    for (int r = 0; r < 8; ++r) {
        long ro = blockBase + r + 8 * halfsel;
        if (ro < (long)n) {
            float* orow = Out + ro * 128 + mlane;
#pragma unroll
            for (int nt = 0; nt < 8; ++nt) orow[nt * 16] = acc[nt][r];
        }
    }
}

// ---------------------------------------------------------------------------
// out[v][:] = bias[:] + t[v][:] * dinv[v]^2     (bias + self-loop message)
// ---------------------------------------------------------------------------
__global__ __launch_bounds__(256)
void init_out_kernel(const float* __restrict__ t, const float* __restrict__ b,
                     const float* __restrict__ dinv, float* __restrict__ out, long total) {
    long idx = (long)blockIdx.x * 256 + threadIdx.x;
    if (idx < total) {
        long row = idx >> 7;
        int  f   = (int)(idx & 127);
        float dv = dinv[row];
        out[idx] = b[f] + t[idx] * dv * dv;
    }
}

// ---------------------------------------------------------------------------
// Edge scatter: one wave per edge, lane handles 4 features (float4).
// out[dst] += t[src] * (dinv[src]*dinv[dst])  via native f32 atomics (L2-bound).
// ---------------------------------------------------------------------------
__global__ __launch_bounds__(256)
void scatter_kernel(const float* __restrict__ t, const float* __restrict__ dinv,
                    const int* __restrict__ src, const int* __restrict__ dst,
                    float* __restrict__ out, int E) {
    long gid  = (long)blockIdx.x * 256 + threadIdx.x;
    long e    = gid >> 5;
    int  lane = threadIdx.x & 31;
    if (e >= (long)E) return;

    int s = __builtin_amdgcn_readfirstlane(src[e]);
    int d = __builtin_amdgcn_readfirstlane(dst[e]);
    float nrm = dinv[s] * dinv[d];

    const float4 v = *(const float4*)(t + (long)s * 128 + lane * 4);
    float* o = out + (long)d * 128 + lane * 4;
    atomic_add_f32(o + 0, v.x * nrm);
    atomic_add_f32(o + 1, v.y * nrm);
    atomic_add_f32(o + 2, v.z * nrm);
    atomic_add_f32(o + 3, v.w * nrm);
}

// ---------------------------------------------------------------------------
// Pooling
// ---------------------------------------------------------------------------
__global__ __launch_bounds__(256) void pool_init_kernel(float* sums, float* cnt) {
    int i = blockIdx.x * 256 + threadIdx.x;
    if (i < 64 * 128) sums[i] = 0.0f;
    if (i < 64)       cnt[i]  = 0.0f;
}

__global__ __launch_bounds__(256)
void pool_accum_kernel(const float* __restrict__ a, const int* __restrict__ batch,
                       float* __restrict__ sums, float* __restrict__ cnt, long total) {
    long idx = (long)blockIdx.x * 256 + threadIdx.x;
    if (idx < total) {
        long v = idx >> 7;
        int  f = (int)(idx & 127);
        int  g = batch[v];
        atomic_add_f32(sums + g * 128 + f, a[idx]);
        if (f == 0) atomic_add_f32(cnt + g, 1.0f);
    }
}

// ---------------------------------------------------------------------------
// Head: pooled = sums/max(cnt,1); y1 = relu(pooled@Wo1+bo1); out = y1@Wo2+bo2
// Single block, all intermediates in LDS (48 KB).
// ---------------------------------------------------------------------------
__global__ __launch_bounds__(256)
void head_kernel(const float* __restrict__ sums, const float* __restrict__ cnt,
                 const float* __restrict__ Wo1, const float* __restrict__ bo1,
                 const float* __restrict__ Wo2, const float* __restrict__ bo2,
                 float* __restrict__ out) {
    __shared__ float pooled[64 * 128];
    __shared__ float y1[64 * 64];
    int tid = threadIdx.x;

    for (int i = tid; i < 64 * 128; i += 256) {
        int g = i >> 7;
        pooled[i] = sums[i] / fmaxf(cnt[g], 1.0f);
    }
    __syncthreads();

    for (int i = tid; i < 64 * 64; i += 256) {
        int g = i >> 6, j = i & 63;
        float acc = bo1[j];
        for (int k = 0; k < 128; ++k) acc += pooled[g * 128 + k] * Wo1[k * 64 + j];
        y1[i] = fmaxf(acc, 0.0f);
    }
    __syncthreads();

    for (int i = tid; i < 64 * 32; i += 256) {
        int g = i >> 5, o = i & 31;
        float acc = bo2[o];
        for (int j = 0; j < 64; ++j) acc += y1[g * 64 + j] * Wo2[j * 32 + o];
        out[i] = acc;
    }
}

// ---------------------------------------------------------------------------
// Launch
// ---------------------------------------------------------------------------
extern "C" void kernel_launch(void* const* d_in, const int* in_sizes, int n_in,
                              void* d_out, int out_size, void* d_ws, size_t ws_size,
                              hipStream_t stream) {
    const float* x   = (const float*)d_in[0];
    const int*   ei  = (const int*)d_in[1];
    const int*   bat = (const int*)d_in[2];
    const float* W1  = (const float*)d_in[3];
    const float* b1  = (const float*)d_in[4];
    const float* W2  = (const float*)d_in[5];
    const float* b2  = (const float*)d_in[6];
    const float* W3  = (const float*)d_in[7];
    const float* b3  = (const float*)d_in[8];
    const float* Wo1 = (const float*)d_in[9];
    const float* bo1 = (const float*)d_in[10];
    const float* Wo2 = (const float*)d_in[11];
    const float* bo2 = (const float*)d_in[12];

    const int n = in_sizes[0] / 128;
    const int E = in_sizes[1] / 2;
    const int* src = ei;
    const int* dst = ei + E;

    char*  ws  = (char*)d_ws;
    size_t off = 0;
    auto alloc = [&](size_t bytes) -> void* {
        void* p = ws + off;
        off = (off + bytes + 255) & ~(size_t)255;
        return p;
    };
    float* t    = (float*)alloc((size_t)n * 128 * sizeof(float));
    float* act  = (float*)alloc((size_t)n * 128 * sizeof(float));
    float* deg  = (float*)alloc((size_t)n * sizeof(float));
    float* dinv = (float*)alloc((size_t)n * sizeof(float));
    float* sums = (float*)alloc(64 * 128 * sizeof(float));
    float* cnt  = (float*)alloc(64 * sizeof(float));

    const dim3 blk(256);
    const long total  = (long)n * 128;
    const int  nb_n   = (n + 255) / 256;
    const int  nb_e   = (E + 255) / 256;
    const int  nb_tot = (int)((total + 255) / 256);
    const int  nb_gm  = (n + 127) / 128;
    const int  nb_sc  = (E + 7) / 8;        // one wave32 per edge

    // degree + normalization
    deg_init_kernel<<<nb_n, blk, 0, stream>>>(deg, n);
    deg_accum_kernel<<<nb_e, blk, 0, stream>>>(dst, deg, E);
    dinv_kernel<<<nb_n, blk, 0, stream>>>(deg, dinv, n);

    // layer 1
    gcn_gemm128_kernel<false><<<nb_gm, blk, 0, stream>>>(x, W1, t, n);
    init_out_kernel<<<nb_tot, blk, 0, stream>>>(t, b1, dinv, act, total);
    scatter_kernel<<<nb_sc, blk, 0, stream>>>(t, dinv, src, dst, act, E);
    // layer 2 (relu fused into GEMM A-load)
    gcn_gemm128_kernel<true><<<nb_gm, blk, 0, stream>>>(act, W2, t, n);
    init_out_kernel<<<nb_tot, blk, 0, stream>>>(t, b2, dinv, act, total);
    scatter_kernel<<<nb_sc, blk, 0, stream>>>(t, dinv, src, dst, act, E);
    // layer 3
    gcn_gemm128_kernel<true><<<nb_gm, blk, 0, stream>>>(act, W3, t, n);
    init_out_kernel<<<nb_tot, blk, 0, stream>>>(t, b3, dinv, act, total);
    scatter_kernel<<<nb_sc, blk, 0, stream>>>(t, dinv, src, dst, act, E);

    // pooling + head
    pool_init_kernel<<<32, blk, 0, stream>>>(sums, cnt);
    pool_accum_kernel<<<nb_tot, blk, 0, stream>>>(act, bat, sums, cnt, total);
    head_kernel<<<1, blk, 0, stream>>>(sums, cnt, Wo1, bo1, Wo2, bo2, (float*)d_out);
}